// CausalGCN_43018392436801
// MI455X (gfx1250) — compile-verified
//
#include <hip/hip_runtime.h>
#include <hip/hip_bf16.h>

typedef __attribute__((ext_vector_type(2))) float v2f;
typedef __attribute__((ext_vector_type(8))) float v8f;

#define DQ 600
#define NBQ 30
#define RQ 8
#define BQ 256
#define KQ 8
#define SQ 128
#define NNODE 2560            // 2*BQ + BQ*KQ
#define WCOLS 5400            // (RQ+1)*DQ  : 8 relation mats + root concatenated

// ---------------- small utility kernels ----------------

__global__ void fill_i32_k(int* p, int v, int n) {
    int i = blockIdx.x * blockDim.x + threadIdx.x;
    if (i < n) p[i] = v;
}

__global__ void concat_x_k(const float* __restrict__ tgt, const float* __restrict__ emo,
                           const float* __restrict__ cau, float* __restrict__ xbuf) {
    int idx = blockIdx.x * blockDim.x + threadIdx.x;
    if (idx >= NNODE * DQ) return;
    int row = idx / DQ, col = idx % DQ;
    float v;
    if (row < BQ)          v = tgt[row * DQ + col];
    else if (row < 2 * BQ) v = emo[(row - BQ) * DQ + col];
    else                   v = cau[(size_t)(row - 2 * BQ) * DQ + col];
    xbuf[idx] = v;
}

// Wcat[i][r*600+o] = sum_b comp[r][b]*basis[b][i][o] ; Wcat[i][4800+o] = root[i][o]
__global__ void build_wcat_k(const float* __restrict__ basis, const float* __restrict__ comp,
                             const float* __restrict__ root, float* __restrict__ wcat) {
    int idx = blockIdx.x * blockDim.x + threadIdx.x;
    if (idx >= DQ * WCOLS) return;
    int i = idx / WCOLS, c = idx % WCOLS;
    int r = c / DQ, o = c % DQ;
    float acc;
    if (r == RQ) {
        acc = root[i * DQ + o];
    } else {
        acc = 0.f;
        #pragma unroll 5
        for (int b = 0; b < NBQ; ++b)
            acc += comp[r * NBQ + b] * basis[(size_t)b * DQ * DQ + (size_t)i * DQ + o];
    }
    wcat[idx] = acc;
}

// ---------------- fp32 WMMA tiled GEMM: C[M x N] = A[M x K] * B[K x N] ----------------
// block tile 64(M) x 128(N), K staged 16 deep in LDS, 8 waves: 4(M) x 2(N),
// each wave owns a 16x64 tile = 4x V_WMMA_F32_16X16X4_F32 accumulators.
// A tile: As[row][k], stride 18 (pad) -> frag pair {k,k+1} is one aligned ds_load_b64,
//         16 lanes' rows hit 16 distinct banks (18*row mod 64 distinct for row 0..15).
// B tile: stored TRANSPOSED Bs[col][k], stride 18 -> frag pair {k,k+1} contiguous,
//         single ds_load_b64 per fragment, no v_mov shuffles in the WMMA loop.
#define APAD 18
#define BPAD 18
__global__ __launch_bounds__(256)
void wmma_gemm_f32_k(const float* __restrict__ A, const float* __restrict__ Bm,
                     float* __restrict__ C, int Kd, int N) {
    __shared__ float As[64][APAD];
    __shared__ float Bs[128][BPAD];

    const int tid  = threadIdx.x;
    const int lane = tid & 31;
    const int wid  = tid >> 5;       // 0..7
    const int wm   = wid >> 1;       // 0..3 (M sub-tile)
    const int wn   = wid & 1;        // 0..1 (N sub-tile)
    const int rowBase = blockIdx.y * 64;
    const int colBase = blockIdx.x * 128;
    const int lrow  = lane & 15;
    const int khalf = (lane >> 4) * 2;   // lanes 16-31 hold K+2,K+3 (ISA 7.12.2)

    v8f c0 = {}, c1 = {}, c2 = {}, c3 = {};

    for (int k0 = 0; k0 < Kd; k0 += 16) {
        // stage A tile (64x16): 512 float2, 2 per thread (Kd is even -> vector-granular tail)
        #pragma unroll
        for (int i = 0; i < 2; ++i) {
            int idx = tid + i * 256;           // 0..511
            int r = idx >> 3, k2 = (idx & 7) * 2;
            v2f v = {0.f, 0.f};
            if (k0 + k2 + 2 <= Kd)
                v = *(const v2f*)&A[(size_t)(rowBase + r) * Kd + k0 + k2];
            *(v2f*)&As[r][k2] = v;
        }
        // stage B tile (16x128) transposed into LDS: float4 global loads (N % 4 == 0),
        // scalar transposed scatter stores
        #pragma unroll
        for (int i = 0; i < 2; ++i) {
            int idx = tid + i * 256;           // 0..511
            int r = idx >> 5, q4 = (idx & 31) * 4;
            float4 v = make_float4(0.f, 0.f, 0.f, 0.f);
            if ((k0 + r < Kd) && (colBase + q4 + 4 <= N))
                v = *(const float4*)&Bm[(size_t)(k0 + r) * N + colBase + q4];
            Bs[q4 + 0][r] = v.x;
            Bs[q4 + 1][r] = v.y;
            Bs[q4 + 2][r] = v.z;
            Bs[q4 + 3][r] = v.w;
        }
        // prefetch next A tile row for this lane (global_prefetch_b8)
        if (k0 + 16 < Kd)
            __builtin_prefetch(&A[(size_t)(rowBase + (tid >> 2)) * Kd + k0 + 16], 0, 0);
        __syncthreads();

        const int cb = wn * 64 + lrow;
        #pragma unroll
        for (int kk = 0; kk < 16; kk += 4) {
            v2f a  = *(const v2f*)&As[wm * 16 + lrow][kk + khalf];
            v2f b0 = *(const v2f*)&Bs[cb +  0][kk + khalf];
            v2f b1 = *(const v2f*)&Bs[cb + 16][kk + khalf];
            v2f b2 = *(const v2f*)&Bs[cb + 32][kk + khalf];
            v2f b3 = *(const v2f*)&Bs[cb + 48][kk + khalf];
            c0 = __builtin_amdgcn_wmma_f32_16x16x4_f32(false, a, false, b0, (short)0, c0, false, false);
            c1 = __builtin_amdgcn_wmma_f32_16x16x4_f32(false, a, false, b1, (short)0, c1, false, false);
            c2 = __builtin_amdgcn_wmma_f32_16x16x4_f32(false, a, false, b2, (short)0, c2, false, false);
            c3 = __builtin_amdgcn_wmma_f32_16x16x4_f32(false, a, false, b3, (short)0, c3, false, false);
        }
        __syncthreads();
    }

    // store C fragments (ISA 7.12.2: VGPR g -> rows g and g+8, N striped over lanes)
    const int rbase = rowBase + wm * 16 + ((lane >> 4) * 8);
    const int cbase = colBase + wn * 64 + (lane & 15);
    #pragma unroll
    for (int g = 0; g < 8; ++g) {
        int row = rbase + g;
        if (cbase +  0 < N) C[(size_t)row * N + cbase +  0] = c0[g];
        if (cbase + 16 < N) C[(size_t)row * N + cbase + 16] = c1[g];
        if (cbase + 32 < N) C[(size_t)row * N + cbase + 32] = c2[g];
        if (cbase + 48 < N) C[(size_t)row * N + cbase + 48] = c3[g];
    }
}

// ---------------- RGCN edge aggregation ----------------

__global__ void rgcn_agg_k(const int* __restrict__ ei, const int* __restrict__ et,
                           const float* __restrict__ XW, float* __restrict__ sums,
                           float* __restrict__ cnt, int E) {
    int e = blockIdx.x;
    int s = ei[e], dn = ei[E + e], t = et[e];
    const float* m = XW + (size_t)s * WCOLS + (size_t)t * DQ;
    float* acc = sums + ((size_t)dn * RQ + t) * DQ;
    for (int d = threadIdx.x; d < DQ; d += blockDim.x)
        atomicAdd(&acc[d], m[d]);
    if (threadIdx.x == 0) atomicAdd(&cnt[dn * RQ + t], 1.0f);
}

__global__ void rgcn_out1_k(const float* __restrict__ sums, const float* __restrict__ cnt,
                            const float* __restrict__ XW, const float* __restrict__ bias,
                            float* __restrict__ out1) {
    int idx = blockIdx.x * blockDim.x + threadIdx.x;
    if (idx >= NNODE * DQ) return;
    int n = idx / DQ, d = idx % DQ;
    float acc = 0.f;
    #pragma unroll
    for (int r = 0; r < RQ; ++r) {
        float c = cnt[n * RQ + r];
        acc += sums[((size_t)n * RQ + r) * DQ + d] / fmaxf(c, 1.0f);
    }
    out1[idx] = acc + XW[(size_t)n * WCOLS + RQ * DQ + d] + bias[d];
}

// ---------------- GAT ----------------

__device__ __forceinline__ int f2ord(float f) { int i = __float_as_int(f); return i ^ ((i >> 31) & 0x7fffffff); }
__device__ __forceinline__ float ord2f(int i) { return __int_as_float(i ^ ((i >> 31) & 0x7fffffff)); }

__global__ void gat_alpha_k(const float* __restrict__ h, const float* __restrict__ att_s,
                            const float* __restrict__ att_d, float* __restrict__ as,
                            float* __restrict__ ad) {
    __shared__ float ss[128], sd[128];
    int n = blockIdx.x;
    float ps = 0.f, pd = 0.f;
    for (int d = threadIdx.x; d < DQ; d += 128) {
        float hv = h[(size_t)n * DQ + d];
        ps += hv * att_s[d];
        pd += hv * att_d[d];
    }
    ss[threadIdx.x] = ps; sd[threadIdx.x] = pd;
    __syncthreads();
    for (int s = 64; s > 0; s >>= 1) {
        if (threadIdx.x < s) { ss[threadIdx.x] += ss[threadIdx.x + s]; sd[threadIdx.x] += sd[threadIdx.x + s]; }
        __syncthreads();
    }
    if (threadIdx.x == 0) { as[n] = ss[0]; ad[n] = sd[0]; }
}

__device__ __forceinline__ void edge_sd(const int* ei, int E, int e, int& s, int& dn) {
    if (e < E) { s = ei[e]; dn = ei[E + e]; } else { s = dn = e - E; }  // self loops appended
}

__global__ void gat_amax_k(const int* __restrict__ ei, const float* __restrict__ as,
                           const float* __restrict__ ad, int* __restrict__ amax, int E) {
    int e = blockIdx.x * blockDim.x + threadIdx.x;
    if (e >= E + NNODE) return;
    int s, dn; edge_sd(ei, E, e, s, dn);
    float a = as[s] + ad[dn];
    a = a > 0.f ? a : 0.2f * a;
    atomicMax(&amax[dn], f2ord(a));
}

__global__ void gat_den_k(const int* __restrict__ ei, const float* __restrict__ as,
                          const float* __restrict__ ad, const int* __restrict__ amax,
                          float* __restrict__ den, int E) {
    int e = blockIdx.x * blockDim.x + threadIdx.x;
    if (e >= E + NNODE) return;
    int s, dn; edge_sd(ei, E, e, s, dn);
    float a = as[s] + ad[dn];
    a = a > 0.f ? a : 0.2f * a;
    atomicAdd(&den[dn], expf(a - ord2f(amax[dn])));
}

__global__ void gat_out2_k(const int* __restrict__ ei, const float* __restrict__ as,
                           const float* __restrict__ ad, const int* __restrict__ amax,
                           const float* __restrict__ den, const float* __restrict__ h,
                           float* __restrict__ out2, int E) {
    int e = blockIdx.x;
    int s, dn; edge_sd(ei, E, e, s, dn);
    float a = as[s] + ad[dn];
    a = a > 0.f ? a : 0.2f * a;
    float coef = expf(a - ord2f(amax[dn])) / den[dn];
    const float* hs = h + (size_t)s * DQ;
    float* od = out2 + (size_t)dn * DQ;
    for (int d = threadIdx.x; d < DQ; d += blockDim.x)
        atomicAdd(&od[d], coef * hs[d]);
}

__global__ void add_bias_k(float* __restrict__ out2, const float* __restrict__ bias) {
    int idx = blockIdx.x * blockDim.x + threadIdx.x;
    if (idx >= NNODE * DQ) return;
    out2[idx] += bias[idx % DQ];
}

// broadcast cause rows over S=128: 629 MB of float4 stores, row staged once in LDS
__global__ __launch_bounds__(256) void bcast_k(const float* __restrict__ out2, float* __restrict__ outf) {
    __shared__ float4 row[DQ / 4];           // 150 float4 per row (600*4B, 16B aligned)
    int c = blockIdx.x;                      // 0..2047 cause nodes
    const float4* src = (const float4*)(out2 + (size_t)(2 * BQ + c) * DQ);
    for (int q = threadIdx.x; q < DQ / 4; q += 256) row[q] = src[q];
    __syncthreads();
    float4* dst = (float4*)(outf + (size_t)c * SQ * DQ);
    for (int idx = threadIdx.x; idx < SQ * (DQ / 4); idx += 256) {
        int t = idx / (DQ / 4), q = idx % (DQ / 4);
        dst[(size_t)t * (DQ / 4) + q] = row[q];
    }
}

// ---------------- launcher ----------------

extern "C" void kernel_launch(void* const* d_in, const int* in_sizes, int n_in,
                              void* d_out, int out_size, void* d_ws, size_t ws_size,
                              hipStream_t stream) {
    (void)n_in; (void)out_size; (void)ws_size;
    const float* target    = (const float*)d_in[0];
    const float* cause     = (const float*)d_in[1];
    const float* emo       = (const float*)d_in[2];
    /* d_in[3] = word_node: UNUSED by the reference -> never read (saves 629 MB) */
    const int*   ei        = (const int*)d_in[4];
    const int*   et        = (const int*)d_in[5];
    const float* basis     = (const float*)d_in[6];
    const float* comp      = (const float*)d_in[7];
    const float* root      = (const float*)d_in[8];
    const float* rgcn_bias = (const float*)d_in[9];
    const float* gat_w     = (const float*)d_in[10];
    const float* att_src   = (const float*)d_in[11];
    const float* att_dst   = (const float*)d_in[12];
    const float* gat_bias  = (const float*)d_in[13];
    const int E = in_sizes[4] / 2;

    float* ws = (float*)d_ws;
    size_t off = 0;
    float* xbuf = ws + off; off += (size_t)NNODE * DQ;
    float* wcat = ws + off; off += (size_t)DQ * WCOLS;
    float* XW   = ws + off; off += (size_t)NNODE * WCOLS;
    float* sums = ws + off; off += (size_t)NNODE * RQ * DQ;
    float* cnt  = ws + off; off += (size_t)NNODE * RQ;
    float* h    = ws + off; off += (size_t)NNODE * DQ;
    float* asrc = ws + off; off += NNODE;
    float* adst = ws + off; off += NNODE;
    float* den  = ws + off; off += NNODE;
    int*   amax = (int*)(ws + off); off += NNODE;

    float* outp      = (float*)d_out;
    float* out_final = outp;                                   // (2048,128,600)
    float* out1      = outp + (size_t)BQ * KQ * SQ * DQ;       // (2560,600)
    float* out2      = out1 + (size_t)NNODE * DQ;              // (2560,600)

    hipMemsetAsync(sums, 0, (size_t)NNODE * RQ * DQ * sizeof(float), stream);
    hipMemsetAsync(cnt,  0, (size_t)NNODE * RQ * sizeof(float), stream);
    hipMemsetAsync(den,  0, NNODE * sizeof(float), stream);
    hipMemsetAsync(out2, 0, (size_t)NNODE * DQ * sizeof(float), stream);
    fill_i32_k<<<(NNODE + 255) / 256, 256, 0, stream>>>(amax, (int)0x80000000, NNODE);

    concat_x_k<<<(NNODE * DQ + 255) / 256, 256, 0, stream>>>(target, emo, cause, xbuf);
    build_wcat_k<<<(DQ * WCOLS + 255) / 256, 256, 0, stream>>>(basis, comp, root, wcat);

    // big GEMM: 2560x600 @ 600x5400 (8 relation mats + root fused)
    dim3 g1((WCOLS + 127) / 128, NNODE / 64);
    wmma_gemm_f32_k<<<g1, 256, 0, stream>>>(xbuf, wcat, XW, DQ, WCOLS);

    rgcn_agg_k<<<E, 128, 0, stream>>>(ei, et, XW, sums, cnt, E);
    rgcn_out1_k<<<(NNODE * DQ + 255) / 256, 256, 0, stream>>>(sums, cnt, XW, rgcn_bias, out1);

    // GAT GEMM: 2560x600 @ 600x600
    dim3 g2((DQ + 127) / 128, NNODE / 64);
    wmma_gemm_f32_k<<<g2, 256, 0, stream>>>(out1, gat_w, h, DQ, DQ);

    gat_alpha_k<<<NNODE, 128, 0, stream>>>(h, att_src, att_dst, asrc, adst);
    gat_amax_k<<<(E + NNODE + 255) / 256, 256, 0, stream>>>(ei, asrc, adst, amax, E);
    gat_den_k<<<(E + NNODE + 255) / 256, 256, 0, stream>>>(ei, asrc, adst, amax, den, E);
    gat_out2_k<<<E + NNODE, 128, 0, stream>>>(ei, asrc, adst, amax, den, h, out2, E);
    add_bias_k<<<(NNODE * DQ + 255) / 256, 256, 0, stream>>>(out2, gat_bias);

    bcast_k<<<BQ * KQ, 256, 0, stream>>>(out2, out_final);
}